// GConv_33964601377480
// MI455X (gfx1250) — compile-verified
//
#include <hip/hip_runtime.h>

#define DIM 128
#define SDIM (DIM + 4)          // padded LDS row: 132 % 64 banks = 4 -> conflict-free A reads
#define NNODES 100000
#define NEDGES 1600000
#define NGRAPHS 512
#define NLAYERS 3
#define ODIM (NLAYERS * DIM)    // 384
#define BN_EPS 1e-5f
#define ROWS 80                 // rows per block: 100000 = 80 * 1250 exactly
#define RT (ROWS / 16)          // 5 row sub-tiles per wave

typedef __attribute__((ext_vector_type(2))) float v2f;
typedef __attribute__((ext_vector_type(8))) float v8f;

// ---------------------------------------------------------------- fill zero
__global__ void gin_zero(float* __restrict__ p, long n) {
    long i = (long)blockIdx.x * blockDim.x + threadIdx.x;
    long stride = (long)gridDim.x * blockDim.x;
    for (; i < n; i += stride) p[i] = 0.0f;
}

// ------------------------------------------- pack W[k][n] -> Wp[k/2][n][k&1]
// so a lane's WMMA B operand {W[k][n], W[k+1][n]} is one aligned 8-byte load.
__global__ __launch_bounds__(256) void gin_packw(const float* __restrict__ W,
                                                 float* __restrict__ Wp) {
    int idx = blockIdx.x * 256 + threadIdx.x;   // 16384 = DIM*DIM
    int k = idx / DIM, n = idx % DIM;
    Wp[(size_t)(k >> 1) * (2 * DIM) + 2 * n + (k & 1)] = W[(size_t)k * DIM + n];
}

// ------------------------------------------------- edge scatter: agg[dst]+=z[src]
// 32 threads per edge, each thread handles 4 contiguous features (float4 gather,
// 4 dword atomics). z and agg are L2-resident (51MB each << 192MB L2).
__global__ __launch_bounds__(256) void gin_scatter(
    const float* __restrict__ z, int zstride,
    const int* __restrict__ src, const int* __restrict__ dst,
    float* __restrict__ agg, int nEdges)
{
    long t = (long)blockIdx.x * blockDim.x + threadIdx.x;
    int e = (int)(t >> 5);
    int c = ((int)t & 31) * 4;
    if (e >= nEdges) return;
    int s = src[e];
    int d = dst[e];
    const float4 v = *(const float4*)(z + (size_t)s * zstride + c);
    float* p = agg + (size_t)d * DIM + c;
    atomicAdd(p + 0, v.x);
    atomicAdd(p + 1, v.y);
    atomicAdd(p + 2, v.z);
    atomicAdd(p + 3, v.w);
}

// ------------------------------------------------- fused GIN MLP via f32 WMMA
// One workgroup = 80 rows; 8 waves, wave w owns output columns [16w,16w+16)
// across 5 row sub-tiles (each B operand feeds 5 WMMAs). in = agg + z staged
// in LDS (padded rows), h1 in LDS, output staged back through sIn for
// coalesced float4 stores in-place over agg (block-private rows -> safe).
// Per-feature BN sums: LDS ds_add_f32 reduce, then 1 global atomic/feature.
__global__ __launch_bounds__(256) void gin_mlp(
    const float* __restrict__ z, int zstride,
    float* __restrict__ hbuf,                 // agg in, h out (stride DIM)
    const v2f* __restrict__ Wp1, const float* __restrict__ b1,
    const v2f* __restrict__ Wp2, const float* __restrict__ b2,
    float* __restrict__ fsum, float* __restrict__ fsq)
{
    __shared__ float sIn[ROWS][SDIM];
    __shared__ float sH1[ROWS][SDIM];
    __shared__ float sSum[DIM];
    __shared__ float sSq[DIM];

    const int tid  = threadIdx.x;
    const int wave = tid >> 5;          // 0..7
    const int lane = tid & 31;
    const int r0   = blockIdx.x * ROWS;

    if (tid < DIM) { sSum[tid] = 0.0f; sSq[tid] = 0.0f; }

    // stage 80x128 input tile: in = agg + z   (256 threads x 10 float4)
    for (int i = tid; i < ROWS * (DIM / 4); i += 256) {
        int m = i >> 5;
        int c = (i & 31) * 4;
        int row = r0 + m;
        const float4 a = *(const float4*)(hbuf + (size_t)row * DIM + c);
        const float4 b = *(const float4*)(z + (size_t)row * zstride + c);
        float4 v;
        v.x = a.x + b.x; v.y = a.y + b.y; v.z = a.z + b.z; v.w = a.w + b.w;
        *(float4*)(&sIn[m][c]) = v;
    }
    __syncthreads();

    const int half = lane >> 4;         // K sub-block select per ISA A layout
    const int l16  = lane & 15;
    const int nCol = wave * 16 + l16;   // output column for B/C/D lanes

    // ---- GEMM1: h1 = relu(in @ W1 + b1); K=128 -> 32 steps x 5 tiles
    v8f acc[RT];
    #pragma unroll
    for (int t = 0; t < RT; t++) acc[t] = (v8f){0.f,0.f,0.f,0.f,0.f,0.f,0.f,0.f};

    for (int k0 = 0; k0 < DIM; k0 += 4) {
        const int ka = k0 + 2 * half;
        const v2f B = Wp1[(size_t)((k0 >> 1) + half) * DIM + nCol];
        #pragma unroll
        for (int t = 0; t < RT; t++) {
            const v2f A = *(const v2f*)(&sIn[t * 16 + l16][ka]);
            acc[t] = __builtin_amdgcn_wmma_f32_16x16x4_f32(
                false, A, false, B, (short)0, acc[t], false, false);
        }
    }
    {
        const float bias = b1[nCol];
        #pragma unroll
        for (int t = 0; t < RT; t++)
            #pragma unroll
            for (int v = 0; v < 8; v++) {
                float h = acc[t][v] + bias;            // C/D: M = v + 8*half
                sH1[t * 16 + v + 8 * half][nCol] = h > 0.0f ? h : 0.0f;
            }
    }
    __syncthreads();

    // ---- GEMM2: h = relu(h1 @ W2 + b2)
    v8f acc2[RT];
    #pragma unroll
    for (int t = 0; t < RT; t++) acc2[t] = (v8f){0.f,0.f,0.f,0.f,0.f,0.f,0.f,0.f};

    for (int k0 = 0; k0 < DIM; k0 += 4) {
        const int ka = k0 + 2 * half;
        const v2f B = Wp2[(size_t)((k0 >> 1) + half) * DIM + nCol];
        #pragma unroll
        for (int t = 0; t < RT; t++) {
            const v2f A = *(const v2f*)(&sH1[t * 16 + l16][ka]);
            acc2[t] = __builtin_amdgcn_wmma_f32_16x16x4_f32(
                false, A, false, B, (short)0, acc2[t], false, false);
        }
    }
    {
        const float bias = b2[nCol];
        float csum = 0.0f, csq = 0.0f;
        #pragma unroll
        for (int t = 0; t < RT; t++)
            #pragma unroll
            for (int v = 0; v < 8; v++) {
                float h = acc2[t][v] + bias;
                h = h > 0.0f ? h : 0.0f;
                sIn[t * 16 + v + 8 * half][nCol] = h;  // restage output tile
                csum += h;
                csq  += h * h;
            }
        atomicAdd(&sSum[nCol], csum);                  // ds_add_f32
        atomicAdd(&sSq[nCol],  csq);
    }
    __syncthreads();

    // coalesced float4 writeback of h over agg rows
    for (int i = tid; i < ROWS * (DIM / 4); i += 256) {
        int m = i >> 5;
        int c = (i & 31) * 4;
        *(float4*)(hbuf + (size_t)(r0 + m) * DIM + c) = *(const float4*)(&sIn[m][c]);
    }
    if (tid < DIM) {
        atomicAdd(&fsum[tid], sSum[tid]);
        atomicAdd(&fsq[tid],  sSq[tid]);
    }
}

// ------------------------------------------------- BN finalize: scale/shift
__global__ void gin_bnfin(const float* __restrict__ fsum, const float* __restrict__ fsq,
                          const float* __restrict__ gamma, const float* __restrict__ beta,
                          float* __restrict__ scale, float* __restrict__ shift, float invN)
{
    int c = threadIdx.x;        // 128 threads
    float mu  = fsum[c] * invN;
    float var = fsq[c] * invN - mu * mu;
    float s   = gamma[c] * rsqrtf(var + BN_EPS);
    scale[c] = s;
    shift[c] = beta[c] - mu * s;
}

// --------------------------- normalize, emit z slice, atomic graph pooling
__global__ __launch_bounds__(256) void gin_norm(
    const float* __restrict__ h,
    const float* __restrict__ scale, const float* __restrict__ shift,
    const int* __restrict__ batch,
    float* __restrict__ zout,   // d_out + l*DIM, stride ODIM
    float* __restrict__ gout,   // d_out + N*ODIM + l*DIM, stride ODIM
    int n)
{
    long t = (long)blockIdx.x * blockDim.x + threadIdx.x;
    int i = (int)(t >> 5);
    int c = ((int)t & 31) * 4;
    if (i >= n) return;
    float4 hv = *(const float4*)(h + (size_t)i * DIM + c);
    float4 sc = *(const float4*)(scale + c);
    float4 sh = *(const float4*)(shift + c);
    float4 zv;
    zv.x = hv.x * sc.x + sh.x;
    zv.y = hv.y * sc.y + sh.y;
    zv.z = hv.z * sc.z + sh.z;
    zv.w = hv.w * sc.w + sh.w;
    *(float4*)(zout + (size_t)i * ODIM + c) = zv;
    int g = batch[i];
    float* gp = gout + (size_t)g * ODIM + c;
    atomicAdd(gp + 0, zv.x);
    atomicAdd(gp + 1, zv.y);
    atomicAdd(gp + 2, zv.z);
    atomicAdd(gp + 3, zv.w);
}

// ---------------------------------------------------------------- launcher
extern "C" void kernel_launch(void* const* d_in, const int* in_sizes, int n_in,
                              void* d_out, int out_size, void* d_ws, size_t ws_size,
                              hipStream_t stream)
{
    (void)in_sizes; (void)n_in; (void)out_size; (void)ws_size;
    const float* x     = (const float*)d_in[0];
    const int*   ei    = (const int*)d_in[1];
    const int*   batch = (const int*)d_in[2];
    const float* W1    = (const float*)d_in[3];
    const float* b1    = (const float*)d_in[4];
    const float* W2    = (const float*)d_in[5];
    const float* b2    = (const float*)d_in[6];
    const float* gamma = (const float*)d_in[7];
    const float* beta  = (const float*)d_in[8];
    float* out = (float*)d_out;

    const int* src = ei;
    const int* dst = ei + NEDGES;

    // workspace: agg[N*DIM] | fsum[128] | fsq[128] | scale[128] | shift[128] | Wp[6*DIM*DIM]
    float* agg   = (float*)d_ws;
    float* fsum  = agg + (size_t)NNODES * DIM;
    float* fsq   = fsum + DIM;
    float* scale = fsq + DIM;
    float* shift = scale + DIM;
    float* wp    = shift + DIM;

    float* zout_base = out;                           // [N, ODIM]
    float* gout_base = out + (size_t)NNODES * ODIM;   // [G, ODIM]

    // zero graph-pool output region once (harness poisons d_out)
    gin_zero<<<512, 256, 0, stream>>>(gout_base, (long)NGRAPHS * ODIM);

    // pre-pack all 6 weight matrices for 8-byte B-operand loads
    for (int l = 0; l < NLAYERS; l++) {
        gin_packw<<<DIM * DIM / 256, 256, 0, stream>>>(
            W1 + (size_t)l * DIM * DIM, wp + (size_t)(2 * l) * DIM * DIM);
        gin_packw<<<DIM * DIM / 256, 256, 0, stream>>>(
            W2 + (size_t)l * DIM * DIM, wp + (size_t)(2 * l + 1) * DIM * DIM);
    }

    const long aggN = (long)NNODES * DIM + 2 * DIM;   // agg + fsum + fsq contiguous
    const int scatterBlocks = (int)(((long)NEDGES * 32 + 255) / 256);
    const int normBlocks    = (int)(((long)NNODES * 32 + 255) / 256);

    for (int l = 0; l < NLAYERS; l++) {
        const float* z = (l == 0) ? x : (zout_base + (size_t)(l - 1) * DIM);
        int zstride    = (l == 0) ? DIM : ODIM;

        gin_zero<<<4096, 256, 0, stream>>>(agg, aggN);

        gin_scatter<<<scatterBlocks, 256, 0, stream>>>(z, zstride, src, dst, agg, NEDGES);

        gin_mlp<<<NNODES / ROWS, 256, 0, stream>>>(
            z, zstride, agg,
            (const v2f*)(wp + (size_t)(2 * l) * DIM * DIM), b1 + (size_t)l * DIM,
            (const v2f*)(wp + (size_t)(2 * l + 1) * DIM * DIM), b2 + (size_t)l * DIM,
            fsum, fsq);

        gin_bnfin<<<1, DIM, 0, stream>>>(fsum, fsq,
                                         gamma + (size_t)l * DIM, beta + (size_t)l * DIM,
                                         scale, shift, 1.0f / NNODES);

        gin_norm<<<normBlocks, 256, 0, stream>>>(
            agg, scale, shift, batch,
            zout_base + (size_t)l * DIM,
            gout_base + (size_t)l * DIM,
            NNODES);
    }
}